// RMAC_55989193671116
// MI455X (gfx1250) — compile-verified
//
#include <hip/hip_runtime.h>

// R-MAC regional max-pool + sum for x[B=32, C=2048, H=32, W=32] f32 -> out[B, C].
// Bandwidth-bound streaming reduction (256 MB @ 23.3 TB/s ~= 11 us floor).
//
// Static geometry (H=W=32, L=(1,2,3)):
//   row ranges (8): [0,32) | [0,21) [5,26) [11,32) | [0,16) [5,21) [10,26) [16,32)
//   col ranges (6): [0,32) | [0,21) [11,32)        | [0,16) [8,24) [16,32)
//   19 regions = 1 + 3x2 + 4x3 (rowRange x colRange); out = sum of region maxes.
//
// Row-range boundaries {0,5,10,11,16,21,26,32} -> 7 disjoint row strips.
// Strips are accumulated as balanced 3-ary max trees so the backend lowers to
// v_max3_f32 (2 ops per 5-row strip per component instead of 4), then 11
// shared max/max3 ops rebuild all 8 row-range maxes.
//
// Wave32 layout: each wave handles 4 planes; lane = (sub = lane>>3, q = lane&7).
// Lane streams its plane's 4-column group as float4 per row -> clause-grouped
// global_load_b128 (512 B payload per wave-instruction). Column segmentation:
// boundaries 8/16/24 are float4-aligned (whole-lane masks on per-lane hmax4);
// only 21/11 need one partial-component select. l3's [0,16)/[16,32) are
// disjoint lane halves -> one 3-shuffle pass produces both segment maxes.

#define PLANE_ELEMS 1024   // 32*32
#define PLANES_PER_WAVE 4
#define WAVES_PER_BLOCK 8

__device__ __forceinline__ float4 vmax4(float4 a, float4 b) {
    return make_float4(fmaxf(a.x, b.x), fmaxf(a.y, b.y),
                       fmaxf(a.z, b.z), fmaxf(a.w, b.w));
}

// Componentwise 3-ary max — shaped for v_max3_f32.
__device__ __forceinline__ float4 vmax43(float4 a, float4 b, float4 c) {
    return make_float4(fmaxf(fmaxf(a.x, b.x), c.x),
                       fmaxf(fmaxf(a.y, b.y), c.y),
                       fmaxf(fmaxf(a.z, b.z), c.z),
                       fmaxf(fmaxf(a.w, b.w), c.w));
}

// Lane-local horizontal max: max3 + max.
__device__ __forceinline__ float hmax4(float4 a) {
    return fmaxf(fmaxf(fmaxf(a.x, a.y), a.z), a.w);
}

// Max across this lane's 8-lane group (xor masks 1,2,4 stay in-group).
__device__ __forceinline__ float grp_max(float t) {
    t = fmaxf(t, __shfl_xor(t, 1, 32));
    t = fmaxf(t, __shfl_xor(t, 2, 32));
    t = fmaxf(t, __shfl_xor(t, 4, 32));
    return t;
}

// Reduce within each 4-lane half of the group, then return (low-half max +
// high-half max) — i.e. segmax[0,16) + segmax[16,32) — uniform over the group.
__device__ __forceinline__ float half_pair_sum(float t) {
    t = fmaxf(t, __shfl_xor(t, 1, 32));
    t = fmaxf(t, __shfl_xor(t, 2, 32));
    return t + __shfl_xor(t, 4, 32);
}

__global__ __launch_bounds__(32 * WAVES_PER_BLOCK)
void rmac_rollup_kernel(const float* __restrict__ x,
                        float* __restrict__ out,
                        int n_planes) {
    const int lane = threadIdx.x & 31;
    const int wave = blockIdx.x * WAVES_PER_BLOCK + (threadIdx.x >> 5);
    const int q    = lane & 7;    // column group (columns 4q..4q+3)
    const int sub  = lane >> 3;   // which of the wave's 4 planes

    const int plane_base = wave * PLANES_PER_WAVE;
    if (plane_base >= n_planes) return;          // wave-uniform guard
    const int plane = plane_base + sub;          // n_planes divisible by 4

    // float4 view: element (r, col) of plane p is float4 index p*256 + r*8 + (col>>2).
    const float4* __restrict__ p =
        reinterpret_cast<const float4*>(x) + (size_t)plane * (PLANE_ELEMS / 4) + q;

#define LD(r) p[(r) * 8]                         // global_load_b128, row r

    // Strip maxes as balanced 3-ary trees (independent operands -> ILP + max3).
    float4 s0 = vmax4(vmax43(LD(0),  LD(1),  LD(2)),  vmax4(LD(3),  LD(4)));  // [0,5)
    float4 s1 = vmax4(vmax43(LD(5),  LD(6),  LD(7)),  vmax4(LD(8),  LD(9)));  // [5,10)
    float4 s2 = LD(10);                                                       // [10,11)
    float4 s3 = vmax4(vmax43(LD(11), LD(12), LD(13)), vmax4(LD(14), LD(15))); // [11,16)
    float4 s4 = vmax4(vmax43(LD(16), LD(17), LD(18)), vmax4(LD(19), LD(20))); // [16,21)
    float4 s5 = vmax4(vmax43(LD(21), LD(22), LD(23)), vmax4(LD(24), LD(25))); // [21,26)
    float4 s6 = vmax4(vmax43(LD(26), LD(27), LD(28)),
                      vmax43(LD(29), LD(30), LD(31)));                        // [26,32)
#undef LD

    // Rebuild the 8 row-range maxes from strips (shared subexpressions).
    float4 A  = vmax43(s1, s2, s3);  // [5,16)
    float4 B  = vmax4(A, s4);        // [5,21)  == R5
    float4 C  = vmax4(s5, s6);       // [21,32)
    float4 R4 = vmax4(s0, A);        // [0,16)
    float4 R1 = vmax4(R4, s4);       // [0,21)
    float4 R5 = B;                   // [5,21)
    float4 R2 = vmax4(B, s5);        // [5,26)
    float4 D  = vmax4(s2, s3);       // [10,16)
    float4 R6 = vmax43(D, s4, s5);   // [10,26)
    float4 R7 = vmax43(s4, s5, s6);  // [16,32)
    float4 R3 = vmax4(s3, R7);       // [11,32)
    float4 R0 = vmax4(R1, C);        // [0,32)

    // Lane-local max over this lane's 4 columns, per row-range.
    const float h0 = hmax4(R0);
    const float h1 = hmax4(R1), h2 = hmax4(R2), h3 = hmax4(R3);
    const float h4 = hmax4(R4), h5 = hmax4(R5), h6 = hmax4(R6), h7 = hmax4(R7);

    const float NEG = -__builtin_inff();

    // l=1: full plane
    float sum = grp_max(h0);

    // l=2 rows {R1,R2,R3} x cols {[0,21), [11,32)}.
    //   [0,21): q<5 whole lane; q==5 only col 20 (.x); else out.
    //   [11,32): q>2 whole lane; q==2 only col 11 (.w); else out.
#define L2_COLS(Rk, hk)                                                  \
    sum += grp_max((q < 5) ? (hk) : ((q == 5) ? (Rk).x : NEG));          \
    sum += grp_max((q > 2) ? (hk) : ((q == 2) ? (Rk).w : NEG));
    L2_COLS(R1, h1)
    L2_COLS(R2, h2)
    L2_COLS(R3, h3)
#undef L2_COLS

    // l=3 rows {R4..R7} x cols {[0,16), [8,24), [16,32)} (all float4-aligned).
    //   [0,16)+[16,32): disjoint lane halves -> one 3-shuffle pass gives both.
    //   [8,24): lanes q=2..5, masked 3-shuffle butterfly.
#define L3_COLS(hk)                                                      \
    sum += half_pair_sum(hk);                                            \
    sum += grp_max((q >= 2 && q < 6) ? (hk) : NEG);
    L3_COLS(h4)
    L3_COLS(h5)
    L3_COLS(h6)
    L3_COLS(h7)
#undef L3_COLS

    if (q == 0) out[plane] = sum;   // plane index == b*C + c (row-major)
}

extern "C" void kernel_launch(void* const* d_in, const int* in_sizes, int n_in,
                              void* d_out, int out_size, void* d_ws, size_t ws_size,
                              hipStream_t stream) {
    (void)n_in; (void)out_size; (void)d_ws; (void)ws_size;

    const float* x = (const float*)d_in[0];
    float* out = (float*)d_out;

    const int total = in_sizes[0];              // B*C*H*W
    const int n_planes = total / PLANE_ELEMS;   // B*C (= 65536)

    const int waves = (n_planes + PLANES_PER_WAVE - 1) / PLANES_PER_WAVE;
    const int blocks = (waves + WAVES_PER_BLOCK - 1) / WAVES_PER_BLOCK;

    rmac_rollup_kernel<<<blocks, 32 * WAVES_PER_BLOCK, 0, stream>>>(x, out, n_planes);
}